// STLDecomposition_180388627310
// MI455X (gfx1250) — compile-verified
//
#include <hip/hip_runtime.h>

// ---------------- problem constants (from reference) ----------------
#define L_LEN   2904
#define F_NUM   207
#define PERIOD  24
#define CYC     121            // L_LEN / PERIOD (odd -> median = rank 60)
#define NF      3              // features per workgroup (207 = 3 * 69)
#define NGROUP  (F_NUM / NF)   // 69
#define WSZ     13             // moving-average window
#define HWIN    6
#define NTHREADS 256
#define NL      (NF * L_LEN)   // 8712 elements per WG tile
#define NMED    (NF * PERIOD)  // 72
#define SMEM_FLOATS (3 * NL + NMED)
#define SMEM_BYTES  (SMEM_FLOATS * sizeof(float))   // ~105 KB (< 320 KB/WGP)

typedef unsigned int       u32;
typedef unsigned long long u64;
typedef __attribute__((ext_vector_type(4))) u32 u32x4;
typedef __attribute__((ext_vector_type(8))) int i32x8;
typedef __attribute__((ext_vector_type(4))) int i32x4;

#if __has_builtin(__builtin_amdgcn_tensor_load_to_lds) && \
    __has_builtin(__builtin_amdgcn_tensor_store_from_lds) && \
    __has_builtin(__builtin_amdgcn_s_wait_tensorcnt)
#define HAVE_TDM 1
#else
#define HAVE_TDM 0
#endif

__device__ __forceinline__ u32 lds_off(const void* p) {
  // Flat LDS pointer: addr[31:0] is the LDS byte offset (aperture in high bits).
  return (u32)(u64)p;
}

#if HAVE_TDM
// 2D TDM transfer: NF consecutive floats per line (features), L_LEN lines with
// stride F_NUM elements between lines. LDS layout = row major: [l*NF + f].
__device__ __forceinline__ void tdm_xfer(bool store, u64 gaddr, u32 laddr) {
  u32x4 g0;
  g0.x = 1u;                                   // count = 1 (valid descriptor)
  g0.y = laddr;                                // lds_addr (bytes)
  g0.z = (u32)gaddr;                           // global_addr[31:0]
  g0.w = ((u32)(gaddr >> 32) & 0x01FFFFFFu)    // global_addr[56:32]
       | 0x80000000u;                          // type = 2 ("image"), bits 127:126
  i32x8 g1;
  g1[0] = 2 << 16;                             // workgroup_mask=0, data_size=2 (4B)
  g1[1] = (int)((u32)NF << 16);                // tensor_dim0[15:0] = NF
  g1[2] = (int)((u32)L_LEN << 16);             // tensor_dim0 hi=0 | tensor_dim1 lo=L
  g1[3] = (int)((u32)NF << 16);                // tensor_dim1 hi=0 | tile_dim0 = NF
  g1[4] = L_LEN;                               // tile_dim1 = L | tile_dim2 = 0
  g1[5] = F_NUM;                               // tensor_dim0_stride = F (elem units)
  g1[6] = 0;                                   // stride hi / dim1_stride lo
  g1[7] = 0;
  i32x4 z4 = {0, 0, 0, 0};
#if defined(__clang_major__) && (__clang_major__ >= 23)
  i32x8 z8 = {0, 0, 0, 0, 0, 0, 0, 0};
  if (store) __builtin_amdgcn_tensor_store_from_lds(g0, g1, z4, z4, z8, 0);
  else       __builtin_amdgcn_tensor_load_to_lds  (g0, g1, z4, z4, z8, 0);
#else
  if (store) __builtin_amdgcn_tensor_store_from_lds(g0, g1, z4, z4, 0);
  else       __builtin_amdgcn_tensor_load_to_lds  (g0, g1, z4, z4, 0);
#endif
}
#endif

extern "C" __global__ void __launch_bounds__(NTHREADS)
stl_decomp_kernel(const float* __restrict__ x,
                  float* __restrict__ o_trend,
                  float* __restrict__ o_seas,
                  float* __restrict__ o_res) {
  extern __shared__ float smem[];
  float* s   = smem;            // [NL]  input series tile, layout s[l*NF + f]
  float* tr  = smem + NL;       // [NL]  trend
  float* db  = smem + 2 * NL;   // [NL]  median staging / residual
  float* med = smem + 3 * NL;   // [NMED] per-(feature,phase) medians

  const int b   = blockIdx.y;
  const int f0  = blockIdx.x * NF;
  const int tid = threadIdx.x;
  const u64 elem_off = (u64)b * L_LEN * F_NUM + (u64)f0;

  // ---- load x[b, :, f0:f0+NF] -> s ----
#if HAVE_TDM
  if (tid == 0) {
    asm volatile("" ::: "memory");
    tdm_xfer(false, (u64)x + 4ull * elem_off, lds_off(s));
    __builtin_amdgcn_s_wait_tensorcnt(0);
    asm volatile("" ::: "memory");
  }
  __syncthreads();
#else
  for (int i = tid; i < NL; i += NTHREADS) {
    int l = i / NF, f = i - l * NF;
    s[i] = x[elem_off + (u64)l * F_NUM + f];
  }
  __syncthreads();
#endif

  // ---- trend0 = MA_13(s) with edge replication ----
  for (int i = tid; i < NL; i += NTHREADS) {
    int l = i / NF, f = i - l * NF;
    float acc = 0.f;
#pragma unroll
    for (int k = -HWIN; k <= HWIN; ++k) {
      int j = l + k;
      j = j < 0 ? 0 : (j > L_LEN - 1 ? L_LEN - 1 : j);
      acc += s[j * NF + f];
    }
    tr[i] = acc * (1.0f / WSZ);
  }

  for (int it = 0; it < 2; ++it) {
    __syncthreads();
    // stage d = s - trend grouped per (feature, phase): db[(f*P+p)*C + c]
    for (int i = tid; i < NL; i += NTHREADS) {
      int l = i / NF, f = i - l * NF;
      int p = l % PERIOD, c = l / PERIOD;
      db[(f * PERIOD + p) * CYC + c] = s[i] - tr[i];
    }
    __syncthreads();
    // median of each 121-row by rank counting; (value,index) tie-break makes the
    // rank-60 element unique and equal to sorted[60] == jnp.median for odd C.
    for (int i = tid; i < NL; i += NTHREADS) {
      int c = i % CYC, fp = i / CYC;
      const float* row = db + fp * CYC;
      float v = row[c];
      int rank = 0;
      for (int j = 0; j < CYC; ++j) {
        float u = row[j];
        rank += (u < v || (u == v && j < c)) ? 1 : 0;
      }
      if (rank == CYC / 2) med[fp] = v;
    }
    __syncthreads();
    // trend = MA_13(s - seasonal), edge replication on the combined signal
    for (int i = tid; i < NL; i += NTHREADS) {
      int l = i / NF, f = i - l * NF;
      float acc = 0.f;
#pragma unroll
      for (int k = -HWIN; k <= HWIN; ++k) {
        int j = l + k;
        j = j < 0 ? 0 : (j > L_LEN - 1 ? L_LEN - 1 : j);
        acc += s[j * NF + f] - med[f * PERIOD + j % PERIOD];
      }
      tr[i] = acc * (1.0f / WSZ);
    }
  }
  __syncthreads();

  // residual -> db, seasonal (broadcast medians) -> s
  for (int i = tid; i < NL; i += NTHREADS) {
    int l = i / NF, f = i - l * NF;
    float se = med[f * PERIOD + l % PERIOD];
    db[i] = s[i] - tr[i] - se;
    s[i]  = se;
  }
  __syncthreads();

  // ---- write trend / seasonal / residual back with same strided pattern ----
#if HAVE_TDM
  if (tid == 0) {
    asm volatile("" ::: "memory");
    tdm_xfer(true, (u64)o_trend + 4ull * elem_off, lds_off(tr));
    tdm_xfer(true, (u64)o_seas  + 4ull * elem_off, lds_off(s));
    tdm_xfer(true, (u64)o_res   + 4ull * elem_off, lds_off(db));
    __builtin_amdgcn_s_wait_tensorcnt(0);
  }
  __syncthreads();
#else
  for (int i = tid; i < NL; i += NTHREADS) {
    int l = i / NF, f = i - l * NF;
    u64 o = elem_off + (u64)l * F_NUM + f;
    o_trend[o] = tr[i];
    o_seas[o]  = s[i];
    o_res[o]   = db[i];
  }
#endif
}

extern "C" void kernel_launch(void* const* d_in, const int* in_sizes, int n_in,
                              void* d_out, int out_size, void* d_ws, size_t ws_size,
                              hipStream_t stream) {
  (void)n_in; (void)out_size; (void)d_ws; (void)ws_size;
  const float* x = (const float*)d_in[0];
  const int per_batch = L_LEN * F_NUM;
  const int B = in_sizes[0] / per_batch;     // 64

  float* o_trend = (float*)d_out;
  float* o_seas  = o_trend + (size_t)B * per_batch;
  float* o_res   = o_seas  + (size_t)B * per_batch;

  static_assert(SMEM_BYTES < 320 * 1024, "LDS tile exceeds WGP capacity");
  // Dynamic LDS > default cap; raise the limit (no-op if already allowed).
  (void)hipFuncSetAttribute((const void*)stl_decomp_kernel,
                            hipFuncAttributeMaxDynamicSharedMemorySize,
                            (int)SMEM_BYTES);

  dim3 grid(NGROUP, B);
  stl_decomp_kernel<<<grid, NTHREADS, SMEM_BYTES, stream>>>(x, o_trend, o_seas, o_res);
}